// SSIM_62680752718194
// MI455X (gfx1250) — compile-verified
//
#include <hip/hip_runtime.h>

// SSIM (7x7 uniform filter, VALID) for [B,1,512,512] f32 pairs -> [B] f32.
// Bandwidth-bound (128 MB @ 23.3 TB/s ~ 5.5us floor). f32 precision kept.
// Vertical 7-tap box sums done on the matrix pipe via V_WMMA_F32_16X16X4_F32
// with a banded 0/1 weight matrix (f32 in / f32 accumulate, no precision loss).

typedef __attribute__((ext_vector_type(2))) float v2f;
typedef __attribute__((ext_vector_type(8))) float v8f;

#define H_IMG   512
#define W_IMG   512
#define OUT_DIM 506          // 512 - 7 + 1
#define WINF    7
#define TILE    32           // output tile per block (32x32)
#define IN_TILE 38           // TILE + WINF - 1
#define XS      39           // padded row stride for input tiles (odd -> no bank conflicts)
#define HS      33           // padded row stride for horizontal-sum tiles

// ---------------------------------------------------------------- kernel 1 --
__global__ __launch_bounds__(256)
void ssim_minmax_kernel(const float* __restrict__ tgt, float* __restrict__ R,
                        double* __restrict__ acc) {
    __shared__ float smn[256];
    __shared__ float smx[256];
    const int b = blockIdx.x;
    const float* p = tgt + (size_t)b * (H_IMG * W_IMG);
    float mn = 3.402823466e38f, mx = -3.402823466e38f;
    for (int i = threadIdx.x; i < H_IMG * W_IMG; i += 256) {
        float v = p[i];
        mn = fminf(mn, v);
        mx = fmaxf(mx, v);
    }
    smn[threadIdx.x] = mn;
    smx[threadIdx.x] = mx;
    __syncthreads();
    for (int s = 128; s > 0; s >>= 1) {
        if (threadIdx.x < s) {
            smn[threadIdx.x] = fminf(smn[threadIdx.x], smn[threadIdx.x + s]);
            smx[threadIdx.x] = fmaxf(smx[threadIdx.x], smx[threadIdx.x + s]);
        }
        __syncthreads();
    }
    if (threadIdx.x == 0) {
        R[b]   = smx[0] - smn[0];
        acc[b] = 0.0;  // reset per-sample accumulator every launch (deterministic)
    }
}

// ---------------------------------------------------------------- kernel 2 --
__global__ __launch_bounds__(128)
void ssim_map_kernel(const float* __restrict__ x, const float* __restrict__ y,
                     const float* __restrict__ Rarr, double* __restrict__ acc) {
    __shared__ float xs[IN_TILE][XS];
    __shared__ float ys[IN_TILE][XS];
    __shared__ float Hc[5][IN_TILE][HS];   // horizontal 7-sums: rows 0..37, cols 0..31

    const int b  = blockIdx.z;
    const int r0 = blockIdx.y * TILE;
    const int c0 = blockIdx.x * TILE;
    const size_t ibase = (size_t)b * (H_IMG * W_IMG);
    const float* xp = x + ibase;
    const float* yp = y + ibase;

    // ---- stage 1: load 38x38 input tiles (edge-clamped; clamped region is
    // only consumed by outputs that get masked out of the mean) -------------
    for (int idx = threadIdx.x; idx < IN_TILE * IN_TILE; idx += 128) {
        const int i  = idx / IN_TILE;
        const int j  = idx - i * IN_TILE;
        const int gr = min(r0 + i, H_IMG - 1);
        const int gc = min(c0 + j, W_IMG - 1);
        const size_t o = (size_t)gr * W_IMG + gc;
        xs[i][j] = xp[o];
        ys[i][j] = yp[o];
    }
    __syncthreads();

    // ---- stage 2: horizontal 7-tap sums for 5 channels ---------------------
    for (int idx = threadIdx.x; idx < IN_TILE * TILE; idx += 128) {
        const int i = idx / TILE;
        const int j = idx - i * TILE;
        float sx = 0.f, sy = 0.f, sxx = 0.f, syy = 0.f, sxy = 0.f;
#pragma unroll
        for (int k = 0; k < WINF; ++k) {
            const float a = xs[i][j + k];
            const float t = ys[i][j + k];
            sx += a;
            sy += t;
            sxx = fmaf(a, a, sxx);
            syy = fmaf(t, t, syy);
            sxy = fmaf(a, t, sxy);
        }
        Hc[0][i][j] = sx;
        Hc[1][i][j] = sy;
        Hc[2][i][j] = sxx;
        Hc[3][i][j] = syy;
        Hc[4][i][j] = sxy;
    }
    __syncthreads();

    // ---- stage 3: vertical 7-tap sums via f32 WMMA -------------------------
    // Each wave32 owns one 16x16 output subtile:
    //   out(16x16) = W(16x24, banded 0/1) * H(24x16), split into six K=4 chunks.
    // A layout (16x4 f32): lanes 0-15 M=lane, VGPRs = K{0,1}; lanes 16-31 = K{2,3}.
    // B layout (4x16 f32): lanes 0-15 N=lane, VGPRs = K{0,1}; lanes 16-31 = K{2,3}.
    const int wave   = threadIdx.x >> 5;   // 0..3
    const int lane   = threadIdx.x & 31;
    const int si     = wave >> 1;          // subtile row (0..1)
    const int sj     = wave & 1;           // subtile col (0..1)
    const int ro     = si * 16;
    const int co     = sj * 16;
    const int mOrN   = lane & 15;          // M for A, N for B/D
    const int hiHalf = lane >> 4;          // selects K pair {0,1} vs {2,3}

    v8f S[5];
#pragma unroll
    for (int ch = 0; ch < 5; ++ch) {
        v8f d = {};
#pragma unroll
        for (int c = 0; c < 6; ++c) {
            const int k0 = 4 * c + 2 * hiHalf;      // global K of .x element
            const int m  = mOrN;
            v2f a;
            a.x = (k0     >= m && k0     <= m + 6) ? 1.0f : 0.0f;
            a.y = (k0 + 1 >= m && k0 + 1 <= m + 6) ? 1.0f : 0.0f;
            // clamp rows 22,23 (weight is 0 there) so we never read junk LDS
            const int rb0 = min(ro + k0,     IN_TILE - 1);
            const int rb1 = min(ro + k0 + 1, IN_TILE - 1);
            v2f bm;
            bm.x = Hc[ch][rb0][co + mOrN];
            bm.y = Hc[ch][rb1][co + mOrN];
            d = __builtin_amdgcn_wmma_f32_16x16x4_f32(
                    /*neg_a=*/false, a, /*neg_b=*/false, bm,
                    /*c_mod=*/(short)0, d, /*reuse_a=*/false, /*reuse_b=*/false);
        }
        S[ch] = d;
    }

    // ---- stage 4: SSIM formula + masked partial mean -----------------------
    const float Rv  = Rarr[b];
    const float C1  = (0.01f * Rv) * (0.01f * Rv);
    const float C2  = (0.03f * Rv) * (0.03f * Rv);
    const float inv = 1.0f / 49.0f;
    const float cvn = 49.0f / 48.0f;     // COV_NORM

    const int gcol = c0 + co + mOrN;
    float psum = 0.f;
#pragma unroll
    for (int v = 0; v < 8; ++v) {
        // D layout: VGPR v -> M = v + 8*hiHalf, N = lane & 15
        const int grow = r0 + ro + v + 8 * hiHalf;
        if (grow < OUT_DIM && gcol < OUT_DIM) {
            const float ux  = S[0][v] * inv;
            const float uy  = S[1][v] * inv;
            const float uxx = S[2][v] * inv;
            const float uyy = S[3][v] * inv;
            const float uxy = S[4][v] * inv;
            const float vx  = cvn * (uxx - ux * ux);
            const float vy  = cvn * (uyy - uy * uy);
            const float vxy = cvn * (uxy - ux * uy);
            const float A1 = 2.f * ux * uy + C1;
            const float A2 = 2.f * vxy + C2;
            const float B1 = ux * ux + uy * uy + C1;
            const float B2 = vx + vy + C2;
            psum += (A1 * A2) / (B1 * B2);
        }
    }
    // wave32 reduction
#pragma unroll
    for (int off = 16; off > 0; off >>= 1)
        psum += __shfl_xor(psum, off, 32);
    if (lane == 0)
        atomicAdd(&acc[b], (double)psum);   // GLOBAL_ATOMIC_ADD_F64
}

// ---------------------------------------------------------------- kernel 3 --
__global__ void ssim_finalize(const double* __restrict__ acc, float* __restrict__ out,
                              int B) {
    const int b = blockIdx.x * blockDim.x + threadIdx.x;
    if (b < B)
        out[b] = (float)(acc[b] * (1.0 / ((double)OUT_DIM * (double)OUT_DIM)));
}

// --------------------------------------------------------------------------
extern "C" void kernel_launch(void* const* d_in, const int* in_sizes, int n_in,
                              void* d_out, int out_size, void* d_ws, size_t ws_size,
                              hipStream_t stream) {
    (void)n_in; (void)ws_size;
    const float* x = (const float*)d_in[0];
    const float* y = (const float*)d_in[1];
    float* out = (float*)d_out;
    const int B = in_sizes[0] / (H_IMG * W_IMG);

    float*  R   = (float*)d_ws;                               // B floats
    double* acc = (double*)((char*)d_ws + (((size_t)B * 4 + 255) & ~(size_t)255));

    ssim_minmax_kernel<<<B, 256, 0, stream>>>(y, R, acc);

    dim3 grid((OUT_DIM + TILE - 1) / TILE, (OUT_DIM + TILE - 1) / TILE, B);
    ssim_map_kernel<<<grid, 128, 0, stream>>>(x, y, R, acc);

    ssim_finalize<<<(B + 63) / 64, 64, 0, stream>>>(acc, out, B);
}